// Deform_29085518528593
// MI455X (gfx1250) — compile-verified
//
#include <hip/hip_runtime.h>
#include <stdint.h>

// Problem constants
#define IMG_H 256
#define IMG_W 256
#define NCH   3
#define BATCH 16
#define NKP1  11
// 16*11*256*256 pixels total
#define TOTAL_PIX (BATCH * NKP1 * IMG_H * IMG_W)

// Tiling: 256-thread block = 8 waves; each wave does 4 double-buffered stages
// of 128 pixels -> 4096 pixels/block -> 2816 blocks exactly.
#define WAVES_PER_BLK 8
#define STAGE_PIX     128            // pixels per wave per stage (128 float2 = 1KB)
#define STAGES        4
#define PIX_PER_WAVE  (STAGE_PIX * STAGES)
#define PIX_PER_BLK   (WAVES_PER_BLK * PIX_PER_WAVE)
#define NUM_BLOCKS    (TOTAL_PIX / PIX_PER_BLK)   // 2816

typedef unsigned int u32x4 __attribute__((ext_vector_type(4)));
typedef int          i32x8 __attribute__((ext_vector_type(8)));
typedef int          i32x4 __attribute__((ext_vector_type(4)));

// 1-D TDM copy: nelem DWORDs from global 'gaddr' to LDS byte offset 'lds_byte'.
// Builds the Tensor DMA Descriptor (D#) per CDNA5 ISA ch.8:
//   group0: count=1 | lds_addr | global_addr[56:0] | type=2
//   group1: data_size=4B (code 2), tensor_dim0=tile_dim0=nelem,
//           tensor_dim1=1, tile_dim1=1 (explicit single row — safe under both
//           readings of "0=unused")
//   group2/3: zero (<=2D tensor -> VADDR2/3 NULL form)
__device__ __forceinline__ void tdm_load_1d(uint32_t lds_byte, uint64_t gaddr,
                                            uint32_t nelem) {
  // Force wave-uniform scalars into SGPRs for the descriptor.
  uint32_t a_lo = (uint32_t)__builtin_amdgcn_readfirstlane((int)(gaddr & 0xFFFFFFFFu));
  uint32_t a_hi = (uint32_t)__builtin_amdgcn_readfirstlane((int)((gaddr >> 32) & 0x1FFFFFFu));
  uint32_t lds  = (uint32_t)__builtin_amdgcn_readfirstlane((int)lds_byte);

  u32x4 g0;
  g0[0] = 1u;                          // count=1 (valid user descriptor)
  g0[1] = lds;                         // lds_addr (bytes)
  g0[2] = a_lo;                        // global_addr[31:0]
  g0[3] = a_hi | (2u << 30);           // global_addr[56:32] | type=2 ("image")

  i32x8 g1;
  g1[0] = (int)(2u << 16);             // wg_mask=0 | data_size=2 (4 bytes)
  g1[1] = (int)((nelem & 0xFFFFu) << 16);          // tensor_dim0[15:0] @ bits 63:48
  g1[2] = (int)(((nelem >> 16) & 0xFFFFu) | (1u << 16)); // tensor_dim0[31:16] | tensor_dim1=1
  g1[3] = (int)((nelem & 0xFFFFu) << 16);          // tile_dim0 @ bits 127:112
  g1[4] = 1;                           // tile_dim1=1 (one row), tile_dim2=0
  g1[5] = (int)nelem;                  // tensor_dim0_stride[31:0]
  g1[6] = 0;                           // tensor_dim0_stride[47:32] | tensor_dim1_stride lo
  g1[7] = 0;

  i32x4 g2 = {0, 0, 0, 0};
  i32x4 g3 = {0, 0, 0, 0};

#if __has_include(<hip/amd_detail/amd_gfx1250_TDM.h>)
  // amdgpu-toolchain (clang-23): 6-arg form
  i32x8 g4 = {0, 0, 0, 0, 0, 0, 0, 0};
  __builtin_amdgcn_tensor_load_to_lds(g0, g1, g2, g3, g4, 0);
#else
  // ROCm 7.2 (clang-22): 5-arg form
  __builtin_amdgcn_tensor_load_to_lds(g0, g1, g2, g3, 0);
#endif
}

__global__ __launch_bounds__(256) void deform_grid_sample_kernel(
    const float* __restrict__ src,      // (256,256,3)
    const float* __restrict__ motions,  // (TOTAL_PIX, 2) flat
    float* __restrict__ out)            // (TOTAL_PIX, 3) flat
{
  // 8 waves x 2 buffers x 128 float2 = 16 KB LDS
  __shared__ float mbuf[WAVES_PER_BLK][2][STAGE_PIX * 2];

  const int wave = (int)(threadIdx.x >> 5);
  const int lane = (int)(threadIdx.x & 31);
  const long long wave_base =
      (long long)blockIdx.x * PIX_PER_BLK + (long long)wave * PIX_PER_WAVE;

  const uint64_t mot_addr = (uint64_t)(uintptr_t)motions;

  // Prologue: stage 0 -> buffer 0 (1 KB, 256 DWORDs)
  {
    uint32_t lds0 = (uint32_t)(uintptr_t)(&mbuf[wave][0][0]);
    tdm_load_1d(lds0, mot_addr + (uint64_t)wave_base * 8ull, STAGE_PIX * 2);
  }

  for (int i = 0; i < STAGES; ++i) {
    // Prefetch next stage into the other buffer, then wait for current stage.
    if (i + 1 < STAGES) {
      uint32_t ldsn = (uint32_t)(uintptr_t)(&mbuf[wave][(i + 1) & 1][0]);
      tdm_load_1d(ldsn,
                  mot_addr + (uint64_t)(wave_base + (long long)(i + 1) * STAGE_PIX) * 8ull,
                  STAGE_PIX * 2);
      __builtin_amdgcn_s_wait_tensorcnt(1);   // current stage's TDM done (in-order)
    } else {
      __builtin_amdgcn_s_wait_tensorcnt(0);   // last stage
    }

    const float* mb = &mbuf[wave][i & 1][0];
    const long long stage_pix = wave_base + (long long)i * STAGE_PIX;

    #pragma unroll
    for (int j = 0; j < STAGE_PIX / 32; ++j) {
      const int li = lane + 32 * j;           // 32 contiguous pixels per j
      const float gx = mb[2 * li + 0];
      const float gy = mb[2 * li + 1];

      // torch-style unnormalize: (g+1)*(256/2) - 0.5 = g*128 + 127.5
      const float x = fmaf(gx, 128.0f, 127.5f);
      const float y = fmaf(gy, 128.0f, 127.5f);
      const float xf = floorf(x);
      const float yf = floorf(y);
      const float fx = x - xf;                // east fraction
      const float fy = y - yf;                // south fraction
      const int x0 = (int)xf, y0 = (int)yf;
      const int x1 = x0 + 1,  y1 = y0 + 1;

      // branchless zero-padding masks
      const float mx0 = (float)((unsigned)x0 < (unsigned)IMG_W);
      const float mx1 = (float)((unsigned)x1 < (unsigned)IMG_W);
      const float my0 = (float)((unsigned)y0 < (unsigned)IMG_H);
      const float my1 = (float)((unsigned)y1 < (unsigned)IMG_H);

      const float w_nw = (1.0f - fy) * (1.0f - fx) * my0 * mx0;
      const float w_ne = (1.0f - fy) * fx          * my0 * mx1;
      const float w_sw = fy          * (1.0f - fx) * my1 * mx0;
      const float w_se = fy          * fx          * my1 * mx1;

      // clamped indices so masked-off corners still read valid memory
      const int x0c = x0 < 0 ? 0 : (x0 > IMG_W - 1 ? IMG_W - 1 : x0);
      const int x1c = x1 < 0 ? 0 : (x1 > IMG_W - 1 ? IMG_W - 1 : x1);
      const int y0c = y0 < 0 ? 0 : (y0 > IMG_H - 1 ? IMG_H - 1 : y0);
      const int y1c = y1 < 0 ? 0 : (y1 > IMG_H - 1 ? IMG_H - 1 : y1);

      // L2-resident gathers (source is 768 KB, stays hot in 192 MB L2)
      const float* pnw = src + (y0c * IMG_W + x0c) * NCH;
      const float* pne = src + (y0c * IMG_W + x1c) * NCH;
      const float* psw = src + (y1c * IMG_W + x0c) * NCH;
      const float* pse = src + (y1c * IMG_W + x1c) * NCH;

      float r0 = w_nw * pnw[0] + w_sw * psw[0] + w_ne * pne[0] + w_se * pse[0];
      float r1 = w_nw * pnw[1] + w_sw * psw[1] + w_ne * pne[1] + w_se * pse[1];
      float r2 = w_nw * pnw[2] + w_sw * psw[2] + w_ne * pne[2] + w_se * pse[2];

      // streaming output: non-temporal so it doesn't evict the source from L2
      float* op = out + (stage_pix + li) * NCH;
      __builtin_nontemporal_store(r0, op + 0);
      __builtin_nontemporal_store(r1, op + 1);
      __builtin_nontemporal_store(r2, op + 2);
    }
  }
}

extern "C" void kernel_launch(void* const* d_in, const int* in_sizes, int n_in,
                              void* d_out, int out_size, void* d_ws, size_t ws_size,
                              hipStream_t stream) {
  const float* src     = (const float*)d_in[0];   // (1,256,256,3) fp32
  const float* motions = (const float*)d_in[1];   // (16,11,256,256,2) fp32
  float* out = (float*)d_out;                     // (16,11,256,256,3) fp32

  deform_grid_sample_kernel<<<dim3(NUM_BLOCKS), dim3(256), 0, stream>>>(
      src, motions, out);
}